// ConditionalSlicedTransport_discrete_37151467110991
// MI455X (gfx1250) — compile-verified
//
#include <hip/hip_runtime.h>
#include <hip/hip_bf16.h>

typedef float v2f __attribute__((ext_vector_type(2)));
typedef float v4f __attribute__((ext_vector_type(4)));
typedef float v8f __attribute__((ext_vector_type(8)));

#define N_CLASS 10
#define NDIM    64
#define NCOMP   32
#define NKNOT   32
#define NTOT    65536

#define ROWS_PER_BLOCK 64     // samples per block
#define TPB            128    // 4 waves (wave32)
#define NWAVES         4
#define SD_STRIDE      68     // data tile LDS stride (bank-conflict-free A reads)
#define SE_STRIDE      36     // E scratch LDS stride

__global__ __launch_bounds__(TPB)
void cst_fused_kernel(const float* __restrict__ data,
                      const int*   __restrict__ label,
                      const float* __restrict__ wT,
                      const float* __restrict__ xx,
                      const float* __restrict__ yy,
                      const float* __restrict__ dl,
                      float* __restrict__ out,
                      float* __restrict__ logj)
{
    __shared__ float sData[ROWS_PER_BLOCK * SD_STRIDE];   // 64x64 tile, padded
    __shared__ v2f   sB1[16 * 2 * 32];                    // wT in WMMA-B lane layout (GEMM1)
    __shared__ v2f   sB2[8 * 4 * 32];                     // wT^T in WMMA-B lane layout (GEMM2)
    __shared__ float sE[NWAVES * 16 * SE_STRIDE];         // E = data1-data0, per-wave scratch
    __shared__ int   sLab[ROWS_PER_BLOCK];

    const int tid  = threadIdx.x;
    const int base = blockIdx.x * ROWS_PER_BLOCK;
    const int lane = tid & 31;
    const int wave = tid >> 5;
    const int hi   = lane >> 4;      // lane half (selects K offset in A/B frags)
    const int l16  = lane & 15;

    // ---- stage data tile: coalesced b128 loads, non-temporal (streamed once) ----
#pragma unroll
    for (int t = 0; t < 8; ++t) {
        int idx = tid + t * TPB;                 // 0..1023
        int row = idx >> 4, c4 = idx & 15;
        v4f v = __builtin_nontemporal_load(
                    (const v4f*)(data + (size_t)(base + row) * NDIM + c4 * 4));
        *(v4f*)(sData + row * SD_STRIDE + c4 * 4) = v;
    }
    // ---- stage B1: B1[K=dim][N=comp] = wT[dim*32+comp], entry (chunk*2+j)*32+lane ----
#pragma unroll
    for (int t = 0; t < 8; ++t) {
        int e = tid + t * TPB;                   // 0..1023
        int ln = e & 31, cj = e >> 5;
        int chunk = cj >> 1, j = cj & 1;
        int krow = 4 * chunk + 2 * (ln >> 4);
        int col  = (ln & 15) + 16 * j;
        v2f b;
        b.x = wT[krow * NCOMP + col];
        b.y = wT[(krow + 1) * NCOMP + col];
        sB1[e] = b;
    }
    // ---- stage B2: B2[K=comp][N=dim] = wT[dim*32+comp], entry (chunk*4+jj)*32+lane ----
#pragma unroll
    for (int t = 0; t < 8; ++t) {
        int e = tid + t * TPB;
        int ln = e & 31, cj = e >> 5;
        int chunk = cj >> 2, jj = cj & 3;
        int n = jj * 16 + (ln & 15);
        int k = 4 * chunk + 2 * (ln >> 4);
        sB2[e] = *(const v2f*)(wT + n * NCOMP + k);   // adjacent comps -> contiguous
    }
    if (tid < ROWS_PER_BLOCK) sLab[tid] = label[base + tid];
    __syncthreads();

    // ================= GEMM1: data0 = data @ wT  (16x32 per wave) =================
    // A fragment (16x4 f32): lane<16 holds (M=lane, K=k..k+1), lane>=16 (M=lane-16, K=k+2..k+3)
    const int rowL = wave * 16 + l16;
    v2f a1[16];
#pragma unroll
    for (int c = 0; c < 16; ++c)
        a1[c] = *(const v2f*)(sData + rowL * SD_STRIDE + 4 * c + 2 * hi);

    v8f c0 = {0.f, 0.f, 0.f, 0.f, 0.f, 0.f, 0.f, 0.f};
    v8f c1 = c0;
#pragma unroll
    for (int c = 0; c < 16; ++c)
        c0 = __builtin_amdgcn_wmma_f32_16x16x4_f32(false, a1[c], false,
                sB1[(c * 2 + 0) * 32 + lane], (short)0, c0, false, false);
#pragma unroll
    for (int c = 0; c < 16; ++c)
        c1 = __builtin_amdgcn_wmma_f32_16x16x4_f32(false, a1[c], false,
                sB1[(c * 2 + 1) * 32 + lane], (short)0, c1, false, false);

    // ============ spline (label-routed), breadth-first search ============
    // C/D frag: vgpr r, lane -> (M = r + 8*hi, N = l16); element e<8: comp l16 (x=c0[e]),
    // e>=8: comp l16+16 (x=c1[e-8]). All 16 searches advance in lock-step rounds so each
    // round is 16 independent L2 loads instead of a serialized dependent chain.
    float* sEw = sE + wave * 16 * SE_STRIDE;
    int kb[16];
    int idx[16];
    float ld[8];

#pragma unroll
    for (int r = 0; r < 8; ++r) {
        int cls = sLab[wave * 16 + r + 8 * hi];
        int k0  = (cls * NCOMP + l16) * NKNOT;
        kb[r]     = k0;
        kb[r + 8] = k0 + 16 * NKNOT;
    }
#define XV(e) (((e) < 8) ? c0[(e) & 7] : c1[(e) & 7])
    // round 1..5: branchless binary search, idx = #knots strictly below x (0..31)
#pragma unroll
    for (int e = 0; e < 16; ++e) idx[e] = (xx[kb[e] + 15] < XV(e)) ? 16 : 0;
#pragma unroll
    for (int e = 0; e < 16; ++e) idx[e] += (xx[kb[e] + idx[e] + 7] < XV(e)) ? 8 : 0;
#pragma unroll
    for (int e = 0; e < 16; ++e) idx[e] += (xx[kb[e] + idx[e] + 3] < XV(e)) ? 4 : 0;
#pragma unroll
    for (int e = 0; e < 16; ++e) idx[e] += (xx[kb[e] + idx[e] + 1] < XV(e)) ? 2 : 0;
#pragma unroll
    for (int e = 0; e < 16; ++e) idx[e] += (xx[kb[e] + idx[e]] < XV(e)) ? 1 : 0;
    // round 6: extend to idx==32 (all knots below x)
#pragma unroll
    for (int e = 0; e < 16; ++e) {
        float t = xx[kb[e] + 31];
        if (idx[e] == 31 && t < XV(e)) idx[e] = 32;
    }

    // gather + RQ math in 2 groups of 8 (caps live VGPRs; 48 independent loads/group)
#pragma unroll
    for (int g = 0; g < 2; ++g) {
        float xk[8], xk1[8], yk[8], yk1[8], dk[8], dk1[8];
#pragma unroll
        for (int r = 0; r < 8; ++r) {
            int e = g * 8 + r;
            int b = idx[e] - 1;
            b = b < 0 ? 0 : (b > NKNOT - 2 ? NKNOT - 2 : b);
            int o = kb[e] + b;
            xk[r] = xx[o]; xk1[r] = xx[o + 1];
            yk[r] = yy[o]; yk1[r] = yy[o + 1];
            dk[r] = dl[o]; dk1[r] = dl[o + 1];
        }
#pragma unroll
        for (int r = 0; r < 8; ++r) {
            int e = g * 8 + r;
            float x   = (g == 0) ? c0[r] : c1[r];
            float w   = xk1[r] - xk[r];
            float s   = (yk1[r] - yk[r]) / w;
            float xi  = fminf(fmaxf((x - xk[r]) / w, 0.0f), 1.0f);
            float xi1 = 1.0f - xi;
            float cc  = xi * xi1;
            float den = s + (dk1[r] + dk[r] - 2.0f * s) * cc;
            float y_in = yk[r] + (yk1[r] - yk[r]) * (s * xi * xi + dk[r] * cc) / den;
            float d_in = s * s * (dk1[r] * xi * xi + 2.0f * s * cc + dk[r] * xi1 * xi1)
                         / (den * den);
            // below -> b==0 so (xk,yk,dk)==(x0,y0,d0); above -> b==30 so knot 31 at [b+1]
            bool below = (idx[e] == 0);
            bool above = (idx[e] == NKNOT);
            float y  = below ? fmaf(x - xk[r], dk[r], yk[r])
                     : (above ? fmaf(x - xk1[r], dk1[r], yk1[r]) : y_in);
            float dd = below ? dk[r] : (above ? dk1[r] : d_in);
            float l  = __logf(dd);
            if (g == 0) ld[r] = l; else ld[r] += l;
            int M = r + 8 * hi;
            sEw[M * SE_STRIDE + l16 + 16 * g] = y - x;   // E = data1 - data0
        }
    }
#undef XV

    // per-sample logj: 2 comps/lane summed in ld[r]; reduce within each 16-lane half
#pragma unroll
    for (int r = 0; r < 8; ++r) {
        float p = ld[r];
        p += __shfl_xor(p, 1, 32);
        p += __shfl_xor(p, 2, 32);
        p += __shfl_xor(p, 4, 32);
        p += __shfl_xor(p, 8, 32);
        if (l16 == 0)
            __builtin_nontemporal_store(p, logj + base + wave * 16 + r + 8 * hi);
    }
    // wave-local LDS RAW fence (DS ops are in-order per wave; block compiler motion)
    asm volatile("s_wait_dscnt 0" ::: "memory");

    // ============ GEMM2: out = data + E @ wT^T  (16x64 per wave) ============
    v2f a2[8];
#pragma unroll
    for (int c = 0; c < 8; ++c)
        a2[c] = *(const v2f*)(sEw + l16 * SE_STRIDE + 4 * c + 2 * hi);

#pragma unroll
    for (int jj = 0; jj < 4; ++jj) {
        v8f acc = {0.f, 0.f, 0.f, 0.f, 0.f, 0.f, 0.f, 0.f};
#pragma unroll
        for (int c = 0; c < 8; ++c)
            acc = __builtin_amdgcn_wmma_f32_16x16x4_f32(false, a2[c], false,
                    sB2[(c * 4 + jj) * 32 + lane], (short)0, acc, false, false);
        int col = jj * 16 + l16;
#pragma unroll
        for (int r = 0; r < 8; ++r) {
            int M    = r + 8 * hi;
            int srow = wave * 16 + M;
            __builtin_nontemporal_store(acc[r] + sData[srow * SD_STRIDE + col],
                                        out + (size_t)(base + srow) * NDIM + col);
        }
    }
}

extern "C" void kernel_launch(void* const* d_in, const int* in_sizes, int n_in,
                              void* d_out, int out_size, void* d_ws, size_t ws_size,
                              hipStream_t stream) {
    const float* data  = (const float*)d_in[0];
    const int*   label = (const int*)  d_in[1];
    const float* wT    = (const float*)d_in[2];
    const float* xx    = (const float*)d_in[3];
    const float* yy    = (const float*)d_in[4];
    const float* dl    = (const float*)d_in[5];
    float* out  = (float*)d_out;
    float* logj = out + (size_t)NTOT * NDIM;   // tuple: out first, then logj

    dim3 grid(NTOT / ROWS_PER_BLOCK);          // 1024 blocks, full tiles only
    cst_fused_kernel<<<grid, TPB, 0, stream>>>(data, label, wT, xx, yy, dl, out, logj);
}